// FastLinearCRF_83434034692284
// MI455X (gfx1250) — compile-verified
//
#include <hip/hip_runtime.h>

typedef float v2f __attribute__((ext_vector_type(2)));
typedef float v4f __attribute__((ext_vector_type(4)));
typedef float v8f __attribute__((ext_vector_type(8)));

#define KLBL 28      // number of labels
#define NB   32      // batch
#define SEQ  512     // sequence length
#define SIDX 25      // START_IDX
#define EIDX 26      // END_IDX
#define LDST 33      // LDS row stride (33 -> conflict-free for 32 rows)

// ---------------------------------------------------------------------------
// Kernel 1: forward partition function (unlabeled score) -> out[0]
// One workgroup, one wave (32 lanes). Sequential over t (inherent dependency),
// per-step 32x28 @ 28x28 exp-space matmul via V_WMMA_F32_16X16X4_F32.
// Lane roles: WMMA fragments during matmul; lane==batch during elementwise.
// ---------------------------------------------------------------------------
__global__ __launch_bounds__(32) void crf_fwd_scan_kernel(
    const float* __restrict__ scores,   // [NB, SEQ, KLBL]
    const int*   __restrict__ lens,     // [NB]
    const float* __restrict__ trans,    // [KLBL, KLBL]
    float* __restrict__ out)
{
  __shared__ float pbuf[NB * LDST];   // rescaled exp-alpha (rows=batch, cols=label, padded)
  __shared__ float dbuf[NB * LDST];   // matmul result

  const int lane = threadIdx.x;
  const int ln   = lane & 15;
  const int kh   = lane >> 4;         // 0: lanes 0-15, 1: lanes 16-31

  // ---- constant B fragments: expT (28x28, zero-padded to 28x32) ----
  // B 4x16 layout: lanes0-15 hold K=4kc+0 (v0) / 4kc+1 (v1); lanes16-31 K=4kc+2 / 4kc+3.
  v2f bfrag[2][7];
  #pragma unroll
  for (int ni = 0; ni < 2; ++ni) {
    #pragma unroll
    for (int kc = 0; kc < 7; ++kc) {
      const int krow = kc * 4 + kh * 2;      // 0..27
      const int col  = ni * 16 + ln;         // 0..31
      float e0 = 0.f, e1 = 0.f;
      if (col < KLBL) {
        e0 = expf(trans[krow       * KLBL + col]);   // exp(-10000) flushes to 0: pads itself
        e1 = expf(trans[(krow + 1) * KLBL + col]);
      }
      bfrag[ni][kc].x = e0;
      bfrag[ni][kc].y = e1;
    }
  }

  // ---- t = 0 initialization (lane = batch) ----
  const int b   = lane;
  const int len = lens[b];
  float c = 0.f, res = 0.f;
  {
    const float* s0 = scores + (b * SEQ) * KLBL;
    float ah[KLBL];
    float m = 0.f;
    #pragma unroll
    for (int j = 0; j < KLBL; ++j) {
      ah[j] = expf(trans[SIDX * KLBL + j] + s0[j]);
      m = fmaxf(m, ah[j]);
    }
    const float inv = 1.f / m;
    #pragma unroll
    for (int j = 0; j < KLBL; ++j) pbuf[b * LDST + j] = ah[j] * inv;
    #pragma unroll
    for (int j = KLBL; j < 32; ++j) pbuf[b * LDST + j] = 0.f;
    c = logf(m);
    if (len == 1) {  // not reachable for these inputs (len >= 256) but kept for generality
      float s = 0.f;
      #pragma unroll
      for (int j = 0; j < KLBL; ++j) s += (ah[j] * inv) * expf(trans[j * KLBL + EIDX]);
      res = c + logf(s);
    }
  }
  __syncthreads();

  // ---- sequential scan ----
  for (int t = 1; t < SEQ; ++t) {
    // prefetch this step's emissions early (independent of the matmul below)
    v4f s4[7];
    const v4f* sp = reinterpret_cast<const v4f*>(scores + (b * SEQ + t) * KLBL); // 112B rows -> 16B aligned
    #pragma unroll
    for (int q = 0; q < 7; ++q) s4[q] = sp[q];

    // D = p @ expT : 32x28 @ 28x32 in f32, 2x2 output tiles, 7 k-chunks of 4
    v8f acc[2][2] = {};
    #pragma unroll
    for (int kc = 0; kc < 7; ++kc) {
      // A 16x4 layout: lanes0-15 row M=ln K=4kc+{0,1}; lanes16-31 K=4kc+{2,3}
      const int base0 = (ln)      * LDST + kc * 4 + kh * 2;
      const int base1 = (16 + ln) * LDST + kc * 4 + kh * 2;
      v2f a0, a1;
      a0.x = pbuf[base0];     a0.y = pbuf[base0 + 1];
      a1.x = pbuf[base1];     a1.y = pbuf[base1 + 1];
      #pragma unroll
      for (int ni = 0; ni < 2; ++ni) {
        acc[0][ni] = __builtin_amdgcn_wmma_f32_16x16x4_f32(
            false, a0, false, bfrag[ni][kc], (short)0, acc[0][ni], false, false);
        acc[1][ni] = __builtin_amdgcn_wmma_f32_16x16x4_f32(
            false, a1, false, bfrag[ni][kc], (short)0, acc[1][ni], false, false);
      }
    }

    // scatter D tiles to LDS (C/D layout: VGPR r -> rows r / r+8)
    #pragma unroll
    for (int mi = 0; mi < 2; ++mi)
      #pragma unroll
      for (int ni = 0; ni < 2; ++ni)
        #pragma unroll
        for (int r = 0; r < 8; ++r) {
          const int row = mi * 16 + kh * 8 + r;
          const int col = ni * 16 + ln;
          dbuf[row * LDST + col] = acc[mi][ni][r];
        }
    __syncthreads();

    // per-batch elementwise tail (lane = batch)
    float ah[KLBL];
    float m = 0.f;
    #pragma unroll
    for (int j = 0; j < KLBL; ++j) {
      const float em = s4[j >> 2][j & 3];
      ah[j] = dbuf[b * LDST + j] * expf(em);
      m = fmaxf(m, ah[j]);
    }
    c += logf(m);
    const float inv = 1.f / m;
    #pragma unroll
    for (int j = 0; j < KLBL; ++j) pbuf[b * LDST + j] = ah[j] * inv;

    if (t == len - 1) {
      float s = 0.f;
      #pragma unroll
      for (int j = 0; j < KLBL; ++j) s += (ah[j] * inv) * expf(trans[j * KLBL + EIDX]);
      res = c + logf(s);
    }
    __syncthreads();
  }

  // deterministic ordered reduction of res over batches
  dbuf[lane] = res;
  __syncthreads();
  if (lane == 0) {
    float u = 0.f;
    #pragma unroll
    for (int j = 0; j < NB; ++j) u += dbuf[j];
    out[0] = u;
  }
}

// ---------------------------------------------------------------------------
// Kernel 2: labeled (gold-path) score -> out[1]. Deterministic block reduction.
// ---------------------------------------------------------------------------
__global__ __launch_bounds__(256) void crf_labeled_kernel(
    const float* __restrict__ scores,
    const int*   __restrict__ lens,
    const int*   __restrict__ tags,
    const float* __restrict__ trans,
    float* __restrict__ out)
{
  __shared__ float red[256];
  float acc = 0.f;
  for (int i = threadIdx.x; i < NB * SEQ; i += 256) {
    const int b = i / SEQ;
    const int t = i - b * SEQ;
    const int len = lens[b];
    if (t < len) {
      const int tg = tags[b * SEQ + t];
      const float em = scores[(b * SEQ + t) * KLBL + tg];
      const float pr = (t == 0) ? trans[SIDX * KLBL + tg]
                                : trans[tags[b * SEQ + t - 1] * KLBL + tg];
      float v = em + pr;
      if (t == len - 1) v += trans[tg * KLBL + EIDX];
      acc += v;
    }
  }
  red[threadIdx.x] = acc;
  __syncthreads();
  for (int s = 128; s > 0; s >>= 1) {
    if (threadIdx.x < s) red[threadIdx.x] += red[threadIdx.x + s];
    __syncthreads();
  }
  if (threadIdx.x == 0) out[1] = red[0];
}

// ---------------------------------------------------------------------------
extern "C" void kernel_launch(void* const* d_in, const int* in_sizes, int n_in,
                              void* d_out, int out_size, void* d_ws, size_t ws_size,
                              hipStream_t stream) {
  const float* scores = (const float*)d_in[0];   // lstm_scores [32,512,28] f32
  const int*   lens   = (const int*)  d_in[1];   // word_seq_lens [32] i32
  const int*   tags   = (const int*)  d_in[2];   // tags [32,512] i32
  // d_in[3] = mask (recomputed from lens; unused)
  const float* trans  = (const float*)d_in[4];   // transition [28,28] f32
  float* out = (float*)d_out;                    // out[0]=unlabeled, out[1]=labeled

  crf_fwd_scan_kernel<<<1, 32, 0, stream>>>(scores, lens, trans, out);
  crf_labeled_kernel<<<1, 256, 0, stream>>>(scores, lens, tags, trans, out);
}